// WindowAttention_14937896256095
// MI455X (gfx1250) — compile-verified
//
#include <hip/hip_runtime.h>
#include <math.h>

// ---------------------------------------------------------------------------
// WindowAttention sparse-PFA fused kernel for MI455X (gfx1250, wave32, WMMA)
//   B=256 windows*batch, N=256 tokens, H=6 heads, HD=32, C=192
//   T_IN=144 incoming sparse width, TOPK=96
// Memory-bound (~1 GB traffic @ 23.3 TB/s); fp32 WMMA for scores + projection.
// K/V staged into LDS by the Tensor Data Mover (TDM) with hardware padding
// (stride 32+1 dwords) for bank-conflict-free WMMA fragment reads.
// ---------------------------------------------------------------------------

typedef __attribute__((ext_vector_type(2))) float v2f;
typedef __attribute__((ext_vector_type(8))) float v8f;
typedef unsigned int u32x4 __attribute__((ext_vector_type(4)));
typedef int i32x4 __attribute__((ext_vector_type(4)));
typedef int i32x8 __attribute__((ext_vector_type(8)));

constexpr int kB   = 256;
constexpr int kN   = 256;
constexpr int kH   = 6;
constexpr int kHD  = 32;
constexpr int kC   = 192;   // H*HD
constexpr int k4C  = 768;
constexpr int kTIN = 144;
constexpr int kTK  = 96;
constexpr float kEPS = 1e-20f;

#if __has_builtin(__builtin_amdgcn_tensor_load_to_lds) && \
    __has_builtin(__builtin_amdgcn_s_wait_tensorcnt)
#define USE_TDM 1
#else
#define USE_TDM 0
#endif

__device__ __forceinline__ v8f wmma_f32x4(v2f a, v2f b, v8f c) {
  // D = A(16x4 f32) * B(4x16 f32) + C(16x16 f32)
  return __builtin_amdgcn_wmma_f32_16x16x4_f32(false, a, false, b, (short)0, c,
                                               false, false);
}

#if USE_TDM
// TDM 2D tile load: 256 rows x 32 f32, global row stride 768 floats,
// LDS destination padded 1 dword per 32 dwords (-> stride-33 layout).
// D# bitfields per CDNA5 ISA.
// Toolchain uses the 6-arg builtin: (g0 u32x4, g1 i32x8, i32x4, i32x4, i32x8, cpol)
__device__ __forceinline__ void tdm_load_kv(unsigned lds_addr,
                                            unsigned long long gaddr) {
  u32x4 g0;
  g0[0] = 1u;                                       // count=1, user descriptor
  g0[1] = lds_addr;                                 // LDS byte address
  g0[2] = (unsigned)(gaddr & 0xffffffffu);          // global_addr[31:0]
  g0[3] = (unsigned)((gaddr >> 32) & 0x01ffffffu)   // global_addr[56:32]
        | (2u << 30);                               // type = 2 ("image")
  i32x8 g1;
  g1[0] = (2 << 16)    // data_size = 4 bytes
        | (1 << 20)    // pad_enable
        | (4 << 22);   // pad_interval code 4 = every 32 dwords (+pad_amount 0 = 1 dword)
  g1[1] = (32 & 0xffff) << 16;    // tensor_dim0 = 32 (bits 63:48)
  g1[2] = (256 & 0xffff) << 16;   // tensor_dim0 hi=0 | tensor_dim1 = 256 lo
  g1[3] = (32 & 0xffff) << 16;    // tensor_dim1 hi=0 | tile_dim0 = 32
  g1[4] = 256;                    // tile_dim1 = 256 | tile_dim2 = 0 (2D)
  g1[5] = 768;                    // tensor_dim0_stride = 768 elements
  g1[6] = 0;                      // stride0 hi | stride1 lo (unused, 2D)
  g1[7] = 0;
  const i32x4 z4 = {0, 0, 0, 0};                  // groups 2/3 unused (2D)
  const i32x8 z8 = {0, 0, 0, 0, 0, 0, 0, 0};
  __builtin_amdgcn_tensor_load_to_lds(g0, g1, z4, z4, z8, 0);
}
#endif

// ===========================================================================
// Kernel A: fused scores (WMMA) -> sparse gather + bias -> softmax ->
//           progressive-focus renorm -> top-96 (+ A*V accumulate) per row.
// One block = one (b, h, 16-row tile). 512 threads = 16 wave32.
// ===========================================================================
__global__ __launch_bounds__(512) void attn_fused(
    const float* __restrict__ qkvp,   // (B, N, 4C)
    const float* __restrict__ pfa_v,  // (B, H, N, T_IN)
    const int*   __restrict__ pfa_i,  // (B, H, N, T_IN)
    const int*   __restrict__ rpi,    // (N, N)
    const float* __restrict__ rpb,    // (NUM_RPB, H)
    float* __restrict__ x_pre,        // (B, N, C)  per-head attn output + lepe
    float* __restrict__ attn_out,     // (B, H, N, TOPK)
    int*   __restrict__ idx_out)      // (B, H, N, TOPK)
{
  __shared__ float sQ[16][kHD + 1];     // scaled q tile
  __shared__ float sK[kN][kHD + 1];     // full K for (b,h), padded (banks)
  __shared__ float sV[kN][kHD + 1];     // full V for (b,h)
  __shared__ float sS[16][kN];          // score tile 16x256
  __shared__ float sTV[16][kTK];        // topk values staging
  __shared__ int   sTI[16][kTK];        // topk indices staging

  const int tid  = threadIdx.x;
  const int lane = tid & 31;
  const int wave = tid >> 5;            // 0..15
  const int it0  = blockIdx.x * 16;     // query-row tile base
  const int h    = blockIdx.y;
  const int b    = blockIdx.z;

  const float scale = 0.17677669529663687f;  // 32^-0.5
  const long  qbase = (long)b * (kN * k4C);

  // ---- stage K, V (256x32) via TDM, scaled Q (16x32) via VALU ------------
#if USE_TDM
  if (wave == 0) {
    const unsigned ldsK = (unsigned)(size_t)&sK[0][0];
    const unsigned ldsV = (unsigned)(size_t)&sV[0][0];
    const unsigned long long gaK =
        (unsigned long long)(size_t)(qkvp + qbase + kC + h * kHD);
    const unsigned long long gaV =
        (unsigned long long)(size_t)(qkvp + qbase + 2 * kC + h * kHD);
    tdm_load_kv(ldsK, gaK);
    tdm_load_kv(ldsV, gaV);
    __builtin_amdgcn_s_wait_tensorcnt(0);
  }
#else
  {
    const int koff = kC + h * kHD;       // k slice within 4C
    const int voff = 2 * kC + h * kHD;   // v slice
    for (int r = tid; r < kN * 8; r += 512) {
      const int n = r >> 3, f = (r & 7) * 4;
      const float4 kv = *(const float4*)(qkvp + qbase + (long)n * k4C + koff + f);
      const float4 vv = *(const float4*)(qkvp + qbase + (long)n * k4C + voff + f);
      sK[n][f] = kv.x; sK[n][f+1] = kv.y; sK[n][f+2] = kv.z; sK[n][f+3] = kv.w;
      sV[n][f] = vv.x; sV[n][f+1] = vv.y; sV[n][f+2] = vv.z; sV[n][f+3] = vv.w;
    }
  }
#endif
  {
    const int qoff = h * kHD;
    for (int r = tid; r < 16 * 8; r += 512) {
      const int m = r >> 3, f = (r & 7) * 4;
      const float4 qv =
          *(const float4*)(qkvp + qbase + (long)(it0 + m) * k4C + qoff + f);
      sQ[m][f]   = qv.x * scale; sQ[m][f+1] = qv.y * scale;
      sQ[m][f+2] = qv.z * scale; sQ[m][f+3] = qv.w * scale;
    }
  }
  __syncthreads();

  // ---- scores: each wave computes one 16x16 tile of the 16x256 row-block --
  {
    const int ct   = wave;          // key column tile
    const int l    = lane & 15;
    const int half = lane >> 4;     // f32 WMMA: lanes 16-31 carry K={2,3}
    v8f acc = {};
    #pragma unroll
    for (int ks = 0; ks < 8; ++ks) {        // K = 32 in steps of 4
      const int kb = ks * 4 + half * 2;
      v2f a, bm;
      a.x  = sQ[l][kb];             a.y  = sQ[l][kb + 1];
      bm.x = sK[ct * 16 + l][kb];   bm.y = sK[ct * 16 + l][kb + 1];
      acc = wmma_f32x4(a, bm, acc);
    }
    #pragma unroll
    for (int r = 0; r < 8; ++r)     // C/D layout: M = r + 8*half, N = l
      sS[r + half * 8][ct * 16 + l] = acc[r];
  }
  __syncthreads();

  // ---- per-row phase: wave w owns row w ----------------------------------
  const int  m     = wave;
  const int  i     = it0 + m;
  const long pbase = (((long)(b * kH + h)) * kN + i) * kTIN;
  const long obase = (((long)(b * kH + h)) * kN + i) * kTK;

  float lv[5];
  float mx = -3.0e38f;
  #pragma unroll
  for (int s = 0; s < 5; ++s) {
    const int t = lane + 32 * s;
    if (t < kTIN) {
      const int   id   = pfa_i[pbase + t];
      const float bias = rpb[rpi[i * kN + id] * kH + h];
      lv[s] = sS[m][id] + bias;
      mx = fmaxf(mx, lv[s]);
    } else {
      lv[s] = -3.0e38f;
    }
  }
  #pragma unroll
  for (int off = 16; off > 0; off >>= 1)
    mx = fmaxf(mx, __shfl_xor(mx, off, 32));

  float sum = 0.f;
  #pragma unroll
  for (int s = 0; s < 5; ++s) {
    const int t = lane + 32 * s;
    lv[s] = (t < kTIN) ? __expf(lv[s] - mx) : 0.f;
    sum += lv[s];
  }
  #pragma unroll
  for (int off = 16; off > 0; off >>= 1)
    sum += __shfl_xor(sum, off, 32);
  const float inv = 1.f / sum;

  float sum2 = 0.f;
  #pragma unroll
  for (int s = 0; s < 5; ++s) {
    const int t = lane + 32 * s;
    if (t < kTIN) { lv[s] = lv[s] * inv * pfa_v[pbase + t]; sum2 += lv[s]; }
  }
  #pragma unroll
  for (int off = 16; off > 0; off >>= 1)
    sum2 += __shfl_xor(sum2, off, 32);
  const float rinv = 1.f / (sum2 + kEPS);
  #pragma unroll
  for (int s = 0; s < 5; ++s) {
    const int t = lane + 32 * s;
    lv[s] = (t < kTIN) ? (lv[s] + kEPS) * rinv : -1.f;   // -1 = consumed/invalid
  }

  // ---- top-96 selection (desc value, asc index tiebreak == lax.top_k),
  //      fused with x[i,:] += val * V[idx,:] (lane = head-dim element) ------
  float acc = 0.f;
  for (int j = 0; j < kTK; ++j) {
    float bv = -2.f; int bt = 1 << 20;
    #pragma unroll
    for (int s = 0; s < 5; ++s)          // local candidates are index-ordered
      if (lv[s] > bv) { bv = lv[s]; bt = lane + 32 * s; }
    #pragma unroll
    for (int off = 16; off > 0; off >>= 1) {
      const float ov = __shfl_xor(bv, off, 32);
      const int   ot = __shfl_xor(bt, off, 32);
      if (ov > bv || (ov == bv && ot < bt)) { bv = ov; bt = ot; }
    }
    if ((bt & 31) == lane) lv[bt >> 5] = -1.f;   // remove winner
    const int nid = pfa_i[pbase + bt];            // uniform -> broadcast load
    if (lane == 0) { sTV[m][j] = bv; sTI[m][j] = nid; }
    acc += bv * sV[nid][lane];
  }

  // + lepe, store per-head output in (b, n, h*HD+d) layout for projection
  acc += qkvp[qbase + (long)i * k4C + 3 * kC + h * kHD + lane];
  x_pre[((long)b * kN + i) * kC + h * kHD + lane] = acc;

  __syncthreads();   // make lane-0 staging visible, then write coalesced
  #pragma unroll
  for (int s = 0; s < 3; ++s) {
    const int j = lane + 32 * s;
    attn_out[obase + j] = sTV[m][j];
    idx_out[obase + j]  = sTI[m][j];
  }
}

// ===========================================================================
// Kernel B: output projection  out = x_pre @ proj_w^T + proj_b
//   (65536 x 192) @ (192 x 192)^T, fp32 WMMA 16x16x4, 48 K-steps.
// One block = one 16-row tile; 12 waves, one 16-col tile each.
// ===========================================================================
__global__ __launch_bounds__(384) void proj_kernel(
    const float* __restrict__ xin,   // (B*N, C)
    const float* __restrict__ pw,    // (C, C) row-major
    const float* __restrict__ pb,    // (C,)
    float* __restrict__ out)         // (B*N, C)
{
  __shared__ float sX[16][kC + 1];   // pad: 192 % 64 == 0 would alias banks

  const int tid  = threadIdx.x;
  const int lane = tid & 31;
  const int wave = tid >> 5;              // 0..11 -> column tile
  const long r0  = (long)blockIdx.x * 16;

  for (int r = tid; r < 16 * (kC / 4); r += 384) {
    const int m = r / (kC / 4), f = (r % (kC / 4)) * 4;
    const float4 xv = *(const float4*)(xin + (r0 + m) * kC + f);
    sX[m][f] = xv.x; sX[m][f+1] = xv.y; sX[m][f+2] = xv.z; sX[m][f+3] = xv.w;
  }
  __syncthreads();

  const int l    = lane & 15;
  const int half = lane >> 4;
  const int j    = wave * 16 + l;         // output column
  v8f acc = {};
  for (int ks = 0; ks < kC / 4; ++ks) {   // 48 K-steps of 4
    const int kb = ks * 4 + half * 2;
    v2f a, bm;
    a.x  = sX[l][kb];              a.y  = sX[l][kb + 1];
    bm.x = pw[(long)j * kC + kb];  bm.y = pw[(long)j * kC + kb + 1];  // L2-hot
    acc = wmma_f32x4(a, bm, acc);
  }
  const float bias = pb[j];
  #pragma unroll
  for (int r = 0; r < 8; ++r)
    out[(r0 + r + half * 8) * kC + j] = acc[r] + bias;
}

// ===========================================================================
extern "C" void kernel_launch(void* const* d_in, const int* in_sizes, int n_in,
                              void* d_out, int out_size, void* d_ws,
                              size_t ws_size, hipStream_t stream) {
  const float* qkvp  = (const float*)d_in[0];
  const float* pfa_v = (const float*)d_in[1];
  const int*   pfa_i = (const int*)d_in[2];
  const int*   rpi   = (const int*)d_in[3];
  const float* rpb   = (const float*)d_in[4];
  const float* pw    = (const float*)d_in[5];
  const float* pb    = (const float*)d_in[6];
  // d_in[7] = shift (==0, unused)

  float* out_x    = (float*)d_out;                                  // (B,N,C)
  float* out_attn = out_x + (size_t)kB * kN * kC;                   // (B,H,N,TK)
  int*   out_idx  = (int*)(out_attn + (size_t)kB * kH * kN * kTK);  // (B,H,N,TK)
  float* x_pre    = (float*)d_ws;                                   // (B,N,C)

  attn_fused<<<dim3(kN / 16, kH, kB), 512, 0, stream>>>(
      qkvp, pfa_v, pfa_i, rpi, rpb, x_pre, out_attn, out_idx);
  proj_kernel<<<dim3((kB * kN) / 16), 384, 0, stream>>>(x_pre, pw, pb, out_x);
}